// LossCDF_14242111553592
// MI455X (gfx1250) — compile-verified
//
#include <hip/hip_runtime.h>
#include <hip/hip_bf16.h>

#define N_BINS 64
#define KNOTS  (N_BINS + 1)   // 65
#define EPS    0.001f

typedef __attribute__((address_space(1))) int as1_int;   // global
typedef __attribute__((address_space(3))) int as3_int;   // LDS

// ---------------------------------------------------------------------------
// Per-scalar bucket search + lerp against LDS-resident knot tables.
// Finds the largest lo in [0,63] with e_t[lo] <= t (e_t[0]==0 <= t always),
// which equals clip(count(e_t <= t) - 1, 0, 63) from the reference.
// ---------------------------------------------------------------------------
__device__ __forceinline__ float lcdf_one(float t, const float* __restrict__ et,
                                          const float* __restrict__ eu) {
  int lo = 0;
  #pragma unroll
  for (int step = 32; step >= 1; step >>= 1) {
    lo += (et[lo + step] <= t) ? step : 0;
  }
  const float et0 = et[lo];
  const float et1 = et[lo + 1];
  const float eu0 = eu[lo];
  const float eu1 = eu[lo + 1];
  return eu0 + (eu1 - eu0) * ((t - et0) / (et1 - et0));
}

// ---------------------------------------------------------------------------
// Kernel B (emitted first so the disasm snippet shows the async prologue):
// stream 131072 elements as float4 (global_load_b128/store_b128), tables
// staged into LDS via gfx1250 async global->LDS loads (ASYNCcnt tracked).
// ---------------------------------------------------------------------------
__global__ __launch_bounds__(256) void lcdf_interp(
    const float4* __restrict__ t4,
    const float* __restrict__ e_t_g, const float* __restrict__ e_u_g,
    float4* __restrict__ out4, int n4,
    const float* __restrict__ t_tail, float* __restrict__ out_tail, int n_tail) {
  __shared__ float et[KNOTS];
  __shared__ float eu[KNOTS];
  const int tid = threadIdx.x;

#if defined(__gfx1250__) && __has_builtin(__builtin_amdgcn_global_load_async_to_lds_b32)
  // CDNA5 async data-mover path: per-lane global->LDS copies, ASYNCcnt tracked.
  if (tid < KNOTS) {
    __builtin_amdgcn_global_load_async_to_lds_b32(
        (as1_int*)(e_t_g + tid), (as3_int*)(et + tid), 0, 0);
  } else if (tid < 2 * KNOTS) {
    __builtin_amdgcn_global_load_async_to_lds_b32(
        (as1_int*)(e_u_g + (tid - KNOTS)), (as3_int*)(eu + (tid - KNOTS)), 0, 0);
  }
  #if __has_builtin(__builtin_amdgcn_s_wait_asynccnt)
  __builtin_amdgcn_s_wait_asynccnt(0);
  #else
  asm volatile("s_wait_asynccnt 0" ::: "memory");
  #endif
#else
  if (tid < KNOTS) {
    et[tid] = e_t_g[tid];
  } else if (tid < 2 * KNOTS) {
    eu[tid - KNOTS] = e_u_g[tid - KNOTS];
  }
#endif
  __syncthreads();

  const int stride = gridDim.x * blockDim.x;
  for (int i = blockIdx.x * blockDim.x + tid; i < n4; i += stride) {
    const float4 v = t4[i];
    float4 r;
    r.x = lcdf_one(v.x, et, eu);
    r.y = lcdf_one(v.y, et, eu);
    r.z = lcdf_one(v.z, et, eu);
    r.w = lcdf_one(v.w, et, eu);
    out4[i] = r;
  }

  // scalar tail (n not divisible by 4) — handled by block 0
  if (blockIdx.x == 0) {
    for (int i = tid; i < n_tail; i += blockDim.x) {
      out_tail[i] = lcdf_one(t_tail[i], et, eu);
    }
  }
}

// ---------------------------------------------------------------------------
// Kernel A: build the 65-knot CDF tables e_t, e_u in workspace.
// One workgroup, 64 threads (2 waves). Cost is negligible (64 exps + scans).
// ---------------------------------------------------------------------------
__global__ __launch_bounds__(64) void lcdf_build_tables(
    const float* __restrict__ l_t, const float* __restrict__ l_u,
    float* __restrict__ e_t, float* __restrict__ e_u) {
  __shared__ float sh[N_BINS];
  const int tid = threadIdx.x;
  const float a = l_t[tid];
  const float b = l_u[tid];

  // max(l_t) for numerically-stable softmax (matches jax.nn.softmax)
  sh[tid] = a;
  __syncthreads();
  #pragma unroll
  for (int off = 32; off > 0; off >>= 1) {
    if (tid < off) sh[tid] = fmaxf(sh[tid], sh[tid + off]);
    __syncthreads();
  }
  const float m = sh[0];
  __syncthreads();

  // softmax(l_t)
  const float ea = __expf(a - m);
  sh[tid] = ea;
  __syncthreads();
  #pragma unroll
  for (int off = 32; off > 0; off >>= 1) {
    if (tid < off) sh[tid] += sh[tid + off];
    __syncthreads();
  }
  const float se = sh[0];
  __syncthreads();

  // w_t = softmax + eps, renormalized
  const float wt_raw = ea / se + EPS;
  sh[tid] = wt_raw;
  __syncthreads();
  #pragma unroll
  for (int off = 32; off > 0; off >>= 1) {
    if (tid < off) sh[tid] += sh[tid + off];
    __syncthreads();
  }
  const float wt = wt_raw / sh[0];
  __syncthreads();

  // w_u = exp(l_u) + eps, normalized
  const float eb = __expf(b) + EPS;
  sh[tid] = eb;
  __syncthreads();
  #pragma unroll
  for (int off = 32; off > 0; off >>= 1) {
    if (tid < off) sh[tid] += sh[tid + off];
    __syncthreads();
  }
  const float wu = eb / sh[0];
  __syncthreads();

  // inclusive scan of wt (Hillis-Steele across 64 threads)
  sh[tid] = wt;
  __syncthreads();
  #pragma unroll
  for (int off = 1; off < 64; off <<= 1) {
    const float add = (tid >= off) ? sh[tid - off] : 0.0f;
    __syncthreads();
    sh[tid] += add;
    __syncthreads();
  }
  if (tid == 0) e_t[0] = 0.0f;
  e_t[tid + 1] = sh[tid];
  __syncthreads();

  // inclusive scan of wu
  sh[tid] = wu;
  __syncthreads();
  #pragma unroll
  for (int off = 1; off < 64; off <<= 1) {
    const float add = (tid >= off) ? sh[tid - off] : 0.0f;
    __syncthreads();
    sh[tid] += add;
    __syncthreads();
  }
  if (tid == 0) e_u[0] = 0.0f;
  e_u[tid + 1] = sh[tid];
}

// ---------------------------------------------------------------------------
// Launch: inputs are (t[32*4096], l_t[64], l_u[64]); output float32[32*4096].
// Workspace layout: e_t at ws[0..64], e_u at ws[128..192] (16B-aligned slots).
// ---------------------------------------------------------------------------
extern "C" void kernel_launch(void* const* d_in, const int* in_sizes, int n_in,
                              void* d_out, int out_size, void* d_ws, size_t ws_size,
                              hipStream_t stream) {
  const float* t   = (const float*)d_in[0];
  const float* l_t = (const float*)d_in[1];
  const float* l_u = (const float*)d_in[2];
  float* out = (float*)d_out;
  float* ws  = (float*)d_ws;

  float* e_t = ws;        // 65 floats
  float* e_u = ws + 128;  // 65 floats

  lcdf_build_tables<<<1, 64, 0, stream>>>(l_t, l_u, e_t, e_u);

  const int n      = out_size;       // 131072 for the reference shapes
  const int n4     = n >> 2;
  const int n_tail = n & 3;

  int blocks = (n4 + 255) / 256;     // 128 blocks for the reference shapes
  if (blocks < 1)   blocks = 1;
  if (blocks > 512) blocks = 512;

  lcdf_interp<<<blocks, 256, 0, stream>>>(
      (const float4*)t, e_t, e_u, (float4*)out, n4,
      t + (n4 << 2), out + (n4 << 2), n_tail);
}